// ISTFTHead_47656957116650
// MI455X (gfx1250) — compile-verified
//
#include <hip/hip_runtime.h>

typedef __attribute__((ext_vector_type(2))) float v2f;
typedef __attribute__((ext_vector_type(8))) float v8f;

#define BATCH  4
#define CIN    22
#define TLEN   500000
#define NBINS  11
#define KSIZE  20
#define STRIDEC 5
#define TRIM   10
#define OUTLEN ((TLEN - 1) * STRIDEC)   // 2499995 per batch

#define TILE_T 128                      // owned time steps per block
#define HALO   3                        // extra steps recomputed at the front
#define NT     (TILE_T + HALO)          // 131
#define TPAD   132
#define ACCN   (STRIDEC * TILE_T + 16)  // 656 (covers max olocal 654)

__global__ __launch_bounds__(256)
void istft_head_kernel(const float* __restrict__ x,
                       const float* __restrict__ wReal,
                       const float* __restrict__ wImag,
                       float* __restrict__ out)
{
    __shared__ float fS[24][TPAD];   // combined features [channel][local t], rows 22..23 = 0
    __shared__ float wS[24][32];     // combined weights  [channel][tap], zero padded
    __shared__ float accS[ACCN];     // overlap-add accumulator in LDS

    const int tid  = threadIdx.x;
    const int lane = tid & 31;
    const int wv   = tid >> 5;                    // wave 0..7
    const int b    = blockIdx.y;
    const int t0   = blockIdx.x * TILE_T;

    // ---- init accumulators & padded feature rows ----
    for (int i = tid; i < ACCN; i += 256) accS[i] = 0.0f;
    for (int i = tid; i < 2 * TPAD; i += 256) fS[22 + i / TPAD][i % TPAD] = 0.0f;

    // ---- combined weight matrix: wReal for c<11, -wImag for 11<=c<22, zeros elsewhere ----
    for (int i = tid; i < 24 * 32; i += 256) {
        int c = i >> 5, k = i & 31;
        float v = 0.0f;
        if (k < KSIZE) {
            if (c < NBINS)           v =  wReal[c * KSIZE + k];
            else if (c < 2 * NBINS)  v = -wImag[(c - NBINS) * KSIZE + k];
        }
        wS[c][k] = v;
    }

    // ---- features: fS[c][tl]=exp(xm)*cos(sin(xp)); fS[c+11][tl]=exp(xm)*sin(sin(xp)) ----
    const size_t xb = (size_t)b * CIN * TLEN;
    for (int i = tid; i < NBINS * NT; i += 256) {
        int c  = i / NT;
        int tl = i - c * NT;
        int t  = t0 - HALO + tl;
        float re = 0.0f, im = 0.0f;
        if (t >= 0 && t < TLEN) {
            float xm    = x[xb + (size_t)c * TLEN + t];
            float xp    = x[xb + (size_t)(c + NBINS) * TLEN + t];
            float mag   = __expf(xm);
            float inner = __sinf(xp);
            re = mag * __cosf(inner);
            im = mag * __sinf(inner);
        }
        fS[c][tl]         = re;
        fS[c + NBINS][tl] = im;
    }

    __syncthreads();

    // ---- B fragments from LDS: B is 4x16 (K x N); K split across lane halves / VGPRs ----
    const int kcol  = lane & 15;            // N = tap column
    const int khalf = (lane >> 4) * 2;      // K sub-index contributed by lane half
    v2f bfr[6][2];
    #pragma unroll
    for (int cbi = 0; cbi < 6; ++cbi) {
        int c = cbi * 4 + khalf;
        bfr[cbi][0].x = wS[c][kcol];      bfr[cbi][0].y = wS[c + 1][kcol];
        bfr[cbi][1].x = wS[c][16 + kcol]; bfr[cbi][1].y = wS[c + 1][16 + kcol];
    }

    // ---- GEMM: P[16 x 20] = f[16 x 24] * W[24 x 20] via 12 x V_WMMA_F32_16X16X4_F32 ----
    // A 16x4 f32 layout: lanes 0-15 M=lane (K0,K1 in v0,v1); lanes 16-31 M=lane-16 (K2,K3)
    const int tlA = HALO + wv * 16 + (lane & 15);
    v8f acc0 = {}; v8f acc1 = {};
    #pragma unroll
    for (int cbi = 0; cbi < 6; ++cbi) {
        int c = cbi * 4 + khalf;
        v2f a;
        a.x = fS[c][tlA];
        a.y = fS[c + 1][tlA];
        acc0 = __builtin_amdgcn_wmma_f32_16x16x4_f32(false, a, false, bfr[cbi][0],
                                                     (short)0, acc0, false, false);
        acc1 = __builtin_amdgcn_wmma_f32_16x16x4_f32(false, a, false, bfr[cbi][1],
                                                     (short)0, acc1, false, false);
    }

    // ---- overlap-add D fragments into LDS (olocal = 5*m + k; collisions when k==k' mod 5) ----
    // C/D layout: VGPR r, lanes 0-15 -> (M=r, N=lane); lanes 16-31 -> (M=r+8, N=lane-16)
    const int mOff = wv * 16 + 8 * (lane >> 4);
    #pragma unroll
    for (int r = 0; r < 8; ++r) {
        int m = mOff + r;
        atomicAdd(&accS[m * STRIDEC + kcol], acc0[r]);
        if (kcol < KSIZE - 16)  // taps 16..19 only; columns 20..31 are zero
            atomicAdd(&accS[m * STRIDEC + 16 + kcol], acc1[r]);
    }

    // ---- halo: 3 leading timesteps spill into owned range; 30 scalar dot products ----
    for (int i = tid; i < HALO * KSIZE; i += 256) {
        int d = i / KSIZE, k = i - d * KSIZE;
        int ol = STRIDEC * (d - HALO) + k;
        if (ol >= 0) {
            float s = 0.0f;
            #pragma unroll
            for (int c = 0; c < 2 * NBINS; ++c) s += fS[c][d] * wS[c][k];
            atomicAdd(&accS[ol], s);
        }
    }

    __syncthreads();

    // ---- exclusive, trimmed store of the owned 640 samples ----
    const size_t ob = (size_t)b * OUTLEN;
    for (int ol = tid; ol < STRIDEC * TILE_T; ol += 256) {
        long j = (long)t0 * STRIDEC + ol - TRIM;
        if (j >= 0 && j < OUTLEN) out[ob + (size_t)j] = accS[ol];
    }
}

extern "C" void kernel_launch(void* const* d_in, const int* in_sizes, int n_in,
                              void* d_out, int out_size, void* d_ws, size_t ws_size,
                              hipStream_t stream) {
    const float* x  = (const float*)d_in[0];   // [4,22,500000]
    const float* wR = (const float*)d_in[1];   // [11,1,20]
    const float* wI = (const float*)d_in[2];   // [11,1,20]
    float* out = (float*)d_out;                // [4,1,2499995]
    dim3 grid((TLEN + TILE_T - 1) / TILE_T, BATCH);
    istft_head_kernel<<<grid, 256, 0, stream>>>(x, wR, wI, out);
}